// RGCN_52158082843452
// MI455X (gfx1250) — compile-verified
//
#include <hip/hip_runtime.h>
#include <hip/hip_bf16.h>

#define NN 50000
#define FIN 128
#define DD 96

typedef __attribute__((ext_vector_type(2))) float v2f;
typedef __attribute__((ext_vector_type(8))) float v8f;

__device__ __forceinline__ v8f wmma_f32(v2f a, v2f b, v8f c) {
  // D = A(16x4) * B(4x16) + C, fp32 in / fp32 accum  -> v_wmma_f32_16x16x4_f32
  return __builtin_amdgcn_wmma_f32_16x16x4_f32(
      /*neg_a=*/false, a, /*neg_b=*/false, b,
      /*c_mod=*/(short)0, c, /*reuse_a=*/false, /*reuse_b=*/false);
}

// ---------------------------------------------------------------- utilities
__global__ __launch_bounds__(256) void zero_kernel(float* __restrict__ p, long n) {
  long i = (long)blockIdx.x * blockDim.x + threadIdx.x;
  if (i < n) p[i] = 0.0f;
}

// etype = argmax(edge_attr, axis=1)  (R==2); degree count per (relation,dst)
__global__ __launch_bounds__(256) void prep_edges_kernel(
    const float* __restrict__ edge_attr, const int* __restrict__ dst,
    int* __restrict__ etype, float* __restrict__ deg, int E) {
  int e = blockIdx.x * blockDim.x + threadIdx.x;
  if (e >= E) return;
  float a0 = edge_attr[2 * e + 0];
  float a1 = edge_attr[2 * e + 1];
  int t = (a1 > a0) ? 1 : 0;   // first-max-wins, matches jnp.argmax
  etype[e] = t;
  atomicAdd(&deg[t * NN + dst[e]], 1.0f);
}

__global__ __launch_bounds__(256) void inv_kernel(const float* __restrict__ deg,
                                                  float* __restrict__ inv, int n) {
  int i = blockIdx.x * blockDim.x + threadIdx.x;
  if (i < n) inv[i] = 1.0f / fmaxf(deg[i], 1.0f);
}

// ------------------------------------------------- h0 = relu(x @ Wf^T + bf)
// One wave computes one 16x16 output tile.  A = x rows, B = Wf^T (= Wf rows).
__global__ __launch_bounds__(32) void linear_relu_kernel(
    const float* __restrict__ x, const float* __restrict__ Wf,
    const float* __restrict__ bf, float* __restrict__ h) {
  const int tile_m = blockIdx.x;            // 0..3124
  const int tile_n = blockIdx.y;            // 0..5
  const int lane   = threadIdx.x;
  const int lm     = lane & 15;
  const int kof    = (lane >> 4) << 1;      // 0 (lanes 0-15) or 2 (lanes 16-31)
  const int row    = tile_m * 16 + lm;
  const int col    = tile_n * 16 + lm;
  const float* xr  = x  + (long)row * FIN;
  const float* wr  = Wf + (long)col * FIN;  // column `col` of Wf^T = row of Wf

  v8f c = {};
#pragma unroll
  for (int k = 0; k < FIN; k += 4) {
    v2f a = { xr[k + kof], xr[k + kof + 1] };
    v2f b = { wr[k + kof], wr[k + kof + 1] };
    c = wmma_f32(a, b, c);
  }

  const float bv    = bf[col];
  const int   mbase = (lane >> 4) << 3;     // rows 0..7 or 8..15 of the tile
#pragma unroll
  for (int v = 0; v < 8; ++v) {
    h[(long)(tile_m * 16 + mbase + v) * DD + col] = fmaxf(c[v] + bv, 0.0f);
  }
}

// --------------------------------------- scatter: agg[etype][dst] += h[src]
// 32 lanes per edge, 3 coalesced float atomics per lane (96 floats/edge).
__global__ __launch_bounds__(256) void scatter_kernel(
    const float* __restrict__ h, const int* __restrict__ src,
    const int* __restrict__ dst, const int* __restrict__ etype,
    float* __restrict__ agg, int E) {
  long t = (long)blockIdx.x * blockDim.x + threadIdx.x;
  int e = (int)(t >> 5);
  int lane = (int)(t & 31);
  if (e >= E) return;
  const int s = src[e];
  const int d = dst[e];
  const int r = etype[e];
  const float* hs = h + (long)s * DD;
  float* ad = agg + ((long)r * NN + d) * DD;
#pragma unroll
  for (int j = 0; j < 3; ++j) {
    int idx = lane + (j << 5);
    atomicAdd(&ad[idx], hs[idx]);
  }
}

// ------- out = relu(h@root + bias + (agg0*inv0)@W0 + (agg1*inv1)@W1)
// One wave per 16x16 output tile; single v8f accumulator across all 3 GEMMs.
__global__ __launch_bounds__(32) void rgcn_update_kernel(
    const float* __restrict__ h, const float* __restrict__ agg,
    const float* __restrict__ inv, const float* __restrict__ root,
    const float* __restrict__ W, const float* __restrict__ bias,
    float* __restrict__ out) {
  const int tile_m = blockIdx.x;            // 0..3124
  const int tile_n = blockIdx.y;            // 0..5
  const int lane   = threadIdx.x;
  const int lm     = lane & 15;
  const int kof    = (lane >> 4) << 1;
  const int row    = tile_m * 16 + lm;
  const int col    = tile_n * 16 + lm;

  v8f c = {};
  {   // h @ root   (root is [K=96][N=96] row-major)
    const float* ar = h + (long)row * DD;
#pragma unroll
    for (int k = 0; k < DD; k += 4) {
      v2f a = { ar[k + kof], ar[k + kof + 1] };
      v2f b = { root[(long)(k + kof) * DD + col],
                root[(long)(k + kof + 1) * DD + col] };
      c = wmma_f32(a, b, c);
    }
  }
#pragma unroll
  for (int r = 0; r < 2; ++r) {   // (agg_r * inv_r) @ W[r], mean folded into A
    const float* ar = agg + ((long)r * NN + row) * DD;
    const float  s  = inv[r * NN + row];
    const float* wr = W + (long)r * DD * DD;
#pragma unroll
    for (int k = 0; k < DD; k += 4) {
      v2f a = { ar[k + kof] * s, ar[k + kof + 1] * s };
      v2f b = { wr[(long)(k + kof) * DD + col],
                wr[(long)(k + kof + 1) * DD + col] };
      c = wmma_f32(a, b, c);
    }
  }

  const float bv    = bias[col];
  const int   mbase = (lane >> 4) << 3;
#pragma unroll
  for (int v = 0; v < 8; ++v) {
    out[(long)(tile_m * 16 + mbase + v) * DD + col] = fmaxf(c[v] + bv, 0.0f);
  }
}

// ---------------------------------------------------------------- launcher
extern "C" void kernel_launch(void* const* d_in, const int* in_sizes, int n_in,
                              void* d_out, int out_size, void* d_ws, size_t ws_size,
                              hipStream_t stream) {
  const float* x         = (const float*)d_in[0];
  const int*   ei        = (const int*)d_in[1];
  const float* edge_attr = (const float*)d_in[2];
  const float* Wf        = (const float*)d_in[3];
  const float* bf        = (const float*)d_in[4];
  const float* W         = (const float*)d_in[5];
  const float* root      = (const float*)d_in[6];
  const float* bias      = (const float*)d_in[7];
  float*       out       = (float*)d_out;

  const int E = in_sizes[1] / 2;            // edge_index is (2, E)
  const int* src = ei;
  const int* dst = ei + E;

  // carve workspace (256B aligned slices)
  char* base = (char*)d_ws;
  auto carve = [&](long bytes) -> char* {
    char* p = base;
    base += (bytes + 255) & ~255L;
    return p;
  };
  int*   etype = (int*)  carve((long)E * 4);
  float* deg   = (float*)carve((long)2 * NN * 4);
  float* inv   = (float*)carve((long)2 * NN * 4);
  float* hA    = (float*)carve((long)NN * DD * 4);
  float* hB    = (float*)carve((long)NN * DD * 4);
  float* agg   = (float*)carve((long)2 * NN * DD * 4);

  const long degN = 2L * NN;
  const long aggN = 2L * NN * DD;

  // --- one-time per call: edge types, mean denominators
  zero_kernel<<<(int)((degN + 255) / 256), 256, 0, stream>>>(deg, degN);
  prep_edges_kernel<<<(E + 255) / 256, 256, 0, stream>>>(edge_attr, dst, etype, deg, E);
  inv_kernel<<<(int)((degN + 255) / 256), 256, 0, stream>>>(deg, inv, (int)degN);

  // --- input projection: h0 = relu(x @ Wf^T + bf)
  dim3 gemm_grid(NN / 16, DD / 16);         // 3125 x 6 tiles
  linear_relu_kernel<<<gemm_grid, 32, 0, stream>>>(x, Wf, bf, hA);

  // --- 3 shared-weight RGCN convolutions
  float* hcur = hA;
  float* hnxt = hB;
  for (int it = 0; it < 3; ++it) {
    zero_kernel<<<(int)((aggN + 255) / 256), 256, 0, stream>>>(agg, aggN);
    long sthreads = (long)E * 32;
    scatter_kernel<<<(int)((sthreads + 255) / 256), 256, 0, stream>>>(
        hcur, src, dst, etype, agg, E);
    float* dstbuf = (it == 2) ? out : hnxt;
    rgcn_update_kernel<<<gemm_grid, 32, 0, stream>>>(
        hcur, agg, inv, root, W, bias, dstbuf);
    float* tmp = hcur; hcur = hnxt; hnxt = tmp;
  }
}